// AttentionModuleLoopback_70781061038436
// MI455X (gfx1250) — compile-verified
//
#include <hip/hip_runtime.h>
#include <math.h>

typedef __attribute__((ext_vector_type(2))) float v2f;
typedef __attribute__((ext_vector_type(4))) float v4f;
typedef __attribute__((ext_vector_type(8))) float v8f;

constexpr int NH = 32;
constexpr int SC = 8192;          // cache length
constexpr int D  = 4096;          // model dim
constexpr int DH = 128;           // head dim
constexpr int QL = 32;            // query length
constexpr int KV = SC + QL;       // 8224
constexpr int NT = KV / 16;       // 514 KV tiles of 16
constexpr int CT = SC / 16;       // 512 tiles come from the cache
constexpr int PART_F = 64 + QL * DH;  // floats per (head,chunk) partial

__device__ __forceinline__ float hmax16(float v) {
  v = fmaxf(v, __shfl_xor(v, 1));
  v = fmaxf(v, __shfl_xor(v, 2));
  v = fmaxf(v, __shfl_xor(v, 4));
  v = fmaxf(v, __shfl_xor(v, 8));
  return v;
}
__device__ __forceinline__ float hsum16(float v) {
  v += __shfl_xor(v, 1);
  v += __shfl_xor(v, 2);
  v += __shfl_xor(v, 4);
  v += __shfl_xor(v, 8);
  return v;
}

// Async global->LDS b128 copy (CDNA5 TDM-adjacent path, tracked by ASYNCcnt).
// ldsoff: wave-relative LDS byte address (low 32 bits of a generic shared ptr,
// per ISA flat->LDS truncation). ga: 64-bit global address.
__device__ __forceinline__ void async_b128(unsigned ldsoff, const void* ga) {
  asm volatile("global_load_async_to_lds_b128 %0, %1, off"
               :: "v"(ldsoff), "v"((unsigned long long)(uintptr_t)ga)
               : "memory");
}
#define WAIT_ASYNC_32() asm volatile("s_wait_asynccnt 0x20" ::: "memory")
#define WAIT_ASYNC_0()  asm volatile("s_wait_asynccnt 0x0"  ::: "memory")

// One wave per block. blockIdx.x = head, blockIdx.y = KV chunk.
__global__ __launch_bounds__(32) void attn_partial(
    const float* __restrict__ q,  const float* __restrict__ kc,
    const float* __restrict__ kn, const float* __restrict__ vc,
    const float* __restrict__ vn, float* __restrict__ out,
    float* __restrict__ ws, int tpc, int finalize)
{
  const int h    = blockIdx.x;
  const int c    = blockIdx.y;
  const int lane = threadIdx.x;
  const int half = lane >> 4;      // 0: lanes 0-15, 1: lanes 16-31
  const int lm   = lane & 15;
  const int hoff = h * DH;

  const int t0 = c * tpc;
  int t1 = t0 + tpc; if (t1 > NT) t1 = NT;

  __shared__ float Qs[QL * DH];         // 16 KB
  __shared__ float Ks[2][16 * DH];      // 2 x 8 KB, double buffered
  __shared__ float Vs[2][16 * DH];      // 2 x 8 KB, double buffered
  __shared__ float Ps[QL * 16];         //  2 KB

  // ---- stage Q asynchronously (32 x 128 f32 = 32 wave-wide b128 async ops) ----
  {
    const float* qb = q + hoff + lane * 4;          // row it, col lane*4
    unsigned lq = (unsigned)(uintptr_t)Qs + lane * 16;
#pragma unroll
    for (int it = 0; it < 32; ++it)
      async_b128(lq + it * (DH * 4), qb + (size_t)it * D);
  }

  // ---- issue one KV tile (16 rows x 128 f32 for K and V) into buffer b ----
  auto issue_tile = [&](int t, int b) {
    const float* ksrc = (t < CT) ? (kc + (size_t)t * 16 * D)
                                 : (kn + (size_t)(t - CT) * 16 * D);
    const float* vsrc = (t < CT) ? (vc + (size_t)t * 16 * D)
                                 : (vn + (size_t)(t - CT) * 16 * D);
    const float* kb = ksrc + hoff + lane * 4;
    const float* vb = vsrc + hoff + lane * 4;
    unsigned lk = (unsigned)(uintptr_t)(Ks[b]) + lane * 16;
    unsigned lv = (unsigned)(uintptr_t)(Vs[b]) + lane * 16;
#pragma unroll
    for (int it = 0; it < 16; ++it) {
      async_b128(lk + it * (DH * 4), kb + (size_t)it * D);
      async_b128(lv + it * (DH * 4), vb + (size_t)it * D);
    }
  };

  // Flash state. Row r = mt*16 + half*8 + j, replicated across each 16-lane half.
  float mrun[2][8], lrun[2][8];
  v8f   acc[2][8];
  const v8f vzero = {0.f,0.f,0.f,0.f,0.f,0.f,0.f,0.f};
#pragma unroll
  for (int mt = 0; mt < 2; ++mt)
#pragma unroll
    for (int j = 0; j < 8; ++j) { mrun[mt][j] = -INFINITY; lrun[mt][j] = 0.f; }
#pragma unroll
  for (int mt = 0; mt < 2; ++mt)
#pragma unroll
    for (int dt = 0; dt < 8; ++dt) acc[mt][dt] = vzero;

  if (t0 < t1) issue_tile(t0, 0);

  for (int t = t0; t < t1; ++t) {
    const int b = (t - t0) & 1;
    if (t + 1 < t1) {
      issue_tile(t + 1, b ^ 1);   // prefetch next tile into other buffer
      WAIT_ASYNC_32();            // <=32 outstanding => current tile (and Q) landed
    } else {
      WAIT_ASYNC_0();
    }
    const float* kbuf = Ks[b];
    const float* vbuf = Vs[b];

    // ---- S = Q * K^T : two 16x16 score tiles, 32 k-steps of 16x16x4 f32 WMMA ----
#pragma unroll
    for (int mt = 0; mt < 2; ++mt) {
      v8f creg = vzero;
#pragma unroll
      for (int kk = 0; kk < 32; ++kk) {
        int kd = kk * 4 + half * 2;
        v2f a = *(const v2f*)(Qs + (mt * 16 + lm) * DH + kd);   // A: M=lm, K=kd..kd+1
        v2f bb = *(const v2f*)(kbuf + lm * DH + kd);            // B: N=lm, K=kd..kd+1
        creg = __builtin_amdgcn_wmma_f32_16x16x4_f32(
            false, a, false, bb, (short)0, creg, false, false);
      }
      // ---- online softmax over this 16-wide slab ----
#pragma unroll
      for (int j = 0; j < 8; ++j) {
        float s    = creg[j];                 // row = mt*16 + half*8 + j, col = lm
        float rmax = hmax16(s);
        float mold = mrun[mt][j];
        float mnew = fmaxf(mold, rmax);
        float p    = __expf(s - mnew);
        float rsum = hsum16(p);
        float sc   = __expf(mold - mnew);
        lrun[mt][j] = lrun[mt][j] * sc + rsum;
        mrun[mt][j] = mnew;
#pragma unroll
        for (int dt = 0; dt < 8; ++dt) acc[mt][dt][j] *= sc;    // rescale running acc
        Ps[(mt * 16 + half * 8 + j) * 16 + lm] = p;             // P tile to LDS
      }
    }

    // ---- acc += P * V : 2 M-tiles x 8 dh-tiles x 4 k-steps of WMMA ----
    // (DS ops are in-order within a wave, so the Ps RAW is safe without a barrier.)
#pragma unroll
    for (int mt = 0; mt < 2; ++mt) {
#pragma unroll
      for (int dt = 0; dt < 8; ++dt) {
        v8f creg = acc[mt][dt];
#pragma unroll
        for (int kk = 0; kk < 4; ++kk) {
          int n0 = kk * 4 + half * 2;
          v2f a = *(const v2f*)(Ps + (mt * 16 + lm) * 16 + n0); // A: M=q row, K=kv pos
          v2f bb;
          bb.x = vbuf[(n0 + 0) * DH + dt * 16 + lm];            // B: K=kv pos, N=dh col
          bb.y = vbuf[(n0 + 1) * DH + dt * 16 + lm];
          creg = __builtin_amdgcn_wmma_f32_16x16x4_f32(
              false, a, false, bb, (short)0, creg, false, false);
        }
        acc[mt][dt] = creg;
      }
    }
  }

  if (finalize) {
#pragma unroll
    for (int mt = 0; mt < 2; ++mt)
#pragma unroll
      for (int j = 0; j < 8; ++j) {
        int row = mt * 16 + half * 8 + j;
        float inv = 1.0f / lrun[mt][j];
#pragma unroll
        for (int dt = 0; dt < 8; ++dt)
          out[(size_t)h * QL * DH + row * DH + dt * 16 + lm] = acc[mt][dt][j] * inv;
      }
  } else {
    float* base = ws + (size_t)(h * gridDim.y + c) * PART_F;
#pragma unroll
    for (int mt = 0; mt < 2; ++mt)
#pragma unroll
      for (int j = 0; j < 8; ++j) {
        int row = mt * 16 + half * 8 + j;
        if (lm == 0) { base[row] = mrun[mt][j]; base[32 + row] = lrun[mt][j]; }
#pragma unroll
        for (int dt = 0; dt < 8; ++dt)
          base[64 + row * DH + dt * 16 + lm] = acc[mt][dt][j];
      }
  }
}

// Merge per-chunk flash partials. grid = NH, block = 128 (one dh column per thread).
__global__ void attn_combine(const float* __restrict__ ws, float* __restrict__ out, int nc)
{
  int h = blockIdx.x;
  int d = threadIdx.x;
  for (int qr = 0; qr < QL; ++qr) {
    float M = -INFINITY;
    for (int c = 0; c < nc; ++c)
      M = fmaxf(M, ws[((size_t)h * nc + c) * PART_F + qr]);
    float L = 0.f, o = 0.f;
    for (int c = 0; c < nc; ++c) {
      const float* b = ws + ((size_t)h * nc + c) * PART_F;
      float w = __expf(b[qr] - M);
      L += b[32 + qr] * w;
      o += b[64 + qr * DH + d] * w;
    }
    out[(size_t)h * QL * DH + qr * DH + d] = o / L;
  }
}

// k_ret / v_ret are exact copies of k_new / v_new.
__global__ void copy_kv(const float* __restrict__ kn, const float* __restrict__ vn,
                        float* __restrict__ out)
{
  int i = blockIdx.x * blockDim.x + threadIdx.x;   // over 32768 float4s
  ((v4f*)(out + QL * D))[i]     = ((const v4f*)kn)[i];
  ((v4f*)(out + 2 * QL * D))[i] = ((const v4f*)vn)[i];
}

extern "C" void kernel_launch(void* const* d_in, const int* in_sizes, int n_in,
                              void* d_out, int out_size, void* d_ws, size_t ws_size,
                              hipStream_t stream)
{
  (void)in_sizes; (void)n_in; (void)out_size;
  const float* q  = (const float*)d_in[0];
  const float* kc = (const float*)d_in[1];
  const float* kn = (const float*)d_in[2];
  const float* vc = (const float*)d_in[3];
  const float* vn = (const float*)d_in[4];
  float* out = (float*)d_out;

  // Pick KV-split factor from available scratch (each (head,chunk) partial = 16640 B).
  const size_t per_head_chunk = (size_t)PART_F * sizeof(float);
  const size_t need_one = (size_t)NH * per_head_chunk;
  int nc = 1, finalize = 1;
  if (ws_size >= 2 * need_one) {
    nc = (int)(ws_size / need_one);
    if (nc > 32) nc = 32;
    finalize = 0;
  }
  int tpc = (NT + nc - 1) / nc;

  attn_partial<<<dim3(NH, nc), dim3(32), 0, stream>>>(
      q, kc, kn, vc, vn, out, (float*)d_ws, tpc, finalize);
  if (!finalize)
    attn_combine<<<dim3(NH), dim3(DH), 0, stream>>>((const float*)d_ws, out, nc);
  copy_kv<<<dim3((QL * D / 4) / 256), dim3(256), 0, stream>>>(kn, vn, out);
}